// TemporalPyramidMamba_4612794876628
// MI455X (gfx1250) — compile-verified
//
#include <hip/hip_runtime.h>
#include <cstdint>
#include <cstddef>

#define D_MODEL 512
#define D_INNER 1024
#define D_STATE 16
#define DT_RANK 32
#define BATCH 4
#define T_LEN 1024

#define USE_ASYNC_LDS 1

typedef __attribute__((ext_vector_type(2))) float v2f;
typedef __attribute__((ext_vector_type(8))) float v8f;

#if __has_builtin(__builtin_amdgcn_wmma_f32_16x16x4_f32)
#define HAVE_WMMA_F32 1
#else
#define HAVE_WMMA_F32 0
#endif

// ---------------------------------------------------------------------------
// Generic WMMA GEMM:  C[M,N] (=|+=) A[M,K] * Bt[N,K]^T  (+ bias[N])
// One wave32 computes a 16x64 tile: one A fragment is reused across 4
// V_WMMA_F32_16X16X4_F32 issues per K-step (4 accumulators).
// A: row stride lda (contiguous K).  Bt: element (n,k) at Bt[n*ldbn+k*ldbk];
// LDBK1=true specializes ldbk==1 so A/B fragment loads become b64.
// ---------------------------------------------------------------------------
template <bool LDBK1>
__global__ __launch_bounds__(256) void gemm_wmma_f32(
    const float* __restrict__ A, int lda,
    const float* __restrict__ Bt, int ldbn, int ldbk,
    const float* __restrict__ bias,
    float* __restrict__ C, int ldc,
    int tiles_m, int tiles_n64, int K, int accumulate)
{
    int wave = (blockIdx.x * (int)blockDim.x + (int)threadIdx.x) >> 5;
    int lane = threadIdx.x & 31;
    int ntiles = tiles_m * tiles_n64;
    if (wave >= ntiles) return;               // wave-uniform: EXEC stays all-1s

    int tm = wave / tiles_n64;
    int tn = wave % tiles_n64;                // 64-wide N tile
    int hi  = (lane >= 16) ? 1 : 0;           // lanes 16-31: K+2 (A/B), M+8 (C)
    int l16 = lane & 15;

    // A 16x4 fragment: lane -> M=l16; v.x/v.y = K, K+1 (+2 for hi half)
    const float* arow = A + (size_t)(tm * 16 + l16) * (size_t)lda + (hi ? 2 : 0);
    // B 4x16 fragments for the four 16-col subtiles
    const float* brow[4];
#pragma unroll
    for (int j = 0; j < 4; ++j)
        brow[j] = Bt + (size_t)(tn * 64 + j * 16 + l16) * (size_t)ldbn
                     + (size_t)(hi ? 2 : 0) * (size_t)(LDBK1 ? 1 : ldbk);

    v8f c0 = {}, c1 = {}, c2 = {}, c3 = {};
#if HAVE_WMMA_F32
    for (int k = 0; k < K; k += 4) {
        if ((k & 63) == 0 && (k + 128) < K)
            __builtin_prefetch(arow + k + 128, 0, 1);   // global_prefetch_b8
        v2f a = *(const v2f*)(arow + k);
        v2f b0, b1, b2, b3;
        if (LDBK1) {
            b0 = *(const v2f*)(brow[0] + k);
            b1 = *(const v2f*)(brow[1] + k);
            b2 = *(const v2f*)(brow[2] + k);
            b3 = *(const v2f*)(brow[3] + k);
        } else {
            size_t k0 = (size_t)k * (size_t)ldbk;
            size_t k1 = (size_t)(k + 1) * (size_t)ldbk;
            b0.x = brow[0][k0]; b0.y = brow[0][k1];
            b1.x = brow[1][k0]; b1.y = brow[1][k1];
            b2.x = brow[2][k0]; b2.y = brow[2][k1];
            b3.x = brow[3][k0]; b3.y = brow[3][k1];
        }
        // 8 args: (neg_a, A, neg_b, B, c_mod, C, reuse_a, reuse_b)
        c0 = __builtin_amdgcn_wmma_f32_16x16x4_f32(false, a, false, b0, (short)0, c0, false, false);
        c1 = __builtin_amdgcn_wmma_f32_16x16x4_f32(false, a, false, b1, (short)0, c1, false, false);
        c2 = __builtin_amdgcn_wmma_f32_16x16x4_f32(false, a, false, b2, (short)0, c2, false, false);
        c3 = __builtin_amdgcn_wmma_f32_16x16x4_f32(false, a, false, b3, (short)0, c3, false, false);
    }
#else
    v8f* cc[4] = {&c0, &c1, &c2, &c3};
    for (int j4 = 0; j4 < 4; ++j4)
        for (int j = 0; j < 8; ++j) {
            const float* ar = A + (size_t)(tm * 16 + j + hi * 8) * (size_t)lda;
            const float* br = Bt + (size_t)(tn * 64 + j4 * 16 + l16) * (size_t)ldbn;
            float s = 0.f;
            for (int k = 0; k < K; ++k) s += ar[k] * br[(size_t)k * (size_t)ldbk];
            (*cc[j4])[j] = s;
        }
#endif

    // C/D layout: VGPR j -> row M=j (+8 for hi half), N = lane%16
    v8f acc[4] = {c0, c1, c2, c3};
#pragma unroll
    for (int j4 = 0; j4 < 4; ++j4) {
        int col = tn * 64 + j4 * 16 + l16;
        float bv = bias ? bias[col] : 0.0f;
#pragma unroll
        for (int j = 0; j < 8; ++j) {
            int row = tm * 16 + j + hi * 8;
            size_t idx = (size_t)row * (size_t)ldc + col;
            float v = acc[j4][j] + bv;
            if (accumulate) C[idx] += v; else C[idx] = v;
        }
    }
}

// ---------------------------------------------------------------------------
// Fused transpose + pyramid pooling: x(B,512,1024) -> token-major U0/U1/U2
// ---------------------------------------------------------------------------
__global__ __launch_bounds__(256) void pool_transpose(
    const float* __restrict__ x,
    float* __restrict__ U0, float* __restrict__ U1, float* __restrict__ U2)
{
    int tid = blockIdx.x * blockDim.x + threadIdx.x;   // BATCH*512*256 threads
    int c  = tid % D_MODEL;
    int r  = tid / D_MODEL;
    int tq = r % (T_LEN / 4);
    int b  = r / (T_LEN / 4);
    const float* xp = x + ((size_t)b * D_MODEL + c) * T_LEN + (size_t)tq * 4;
    float a0 = xp[0], a1 = xp[1], a2 = xp[2], a3 = xp[3];
    size_t u0 = ((size_t)b * T_LEN + (size_t)tq * 4) * D_MODEL + c;
    U0[u0]               = a0;
    U0[u0 + D_MODEL]     = a1;
    U0[u0 + 2 * D_MODEL] = a2;
    U0[u0 + 3 * D_MODEL] = a3;
    size_t u1 = ((size_t)b * (T_LEN / 2) + (size_t)tq * 2) * D_MODEL + c;
    U1[u1]           = 0.5f * (a0 + a1);
    U1[u1 + D_MODEL] = 0.5f * (a2 + a3);
    U2[((size_t)b * (T_LEN / 4) + tq) * D_MODEL + c] = 0.25f * (a0 + a1 + a2 + a3);
}

// ---------------------------------------------------------------------------
// Depthwise causal conv (width 4) + bias + SiLU over x-half of XZ
// ---------------------------------------------------------------------------
__global__ __launch_bounds__(256) void conv_silu(
    const float* __restrict__ XZ, const float* __restrict__ w,
    const float* __restrict__ bias, float* __restrict__ XA, int T)
{
    int tid = blockIdx.x * blockDim.x + threadIdx.x;   // M * D_INNER
    int d = tid % D_INNER;
    int m = tid / D_INNER;
    int t = m % T;
    const float* wd = w + (size_t)d * 4;
    float acc = bias[d];
#pragma unroll
    for (int j = 0; j < 4; ++j) {
        int tt = t - 3 + j;
        if (tt >= 0) acc += wd[j] * XZ[(size_t)(m - 3 + j) * (2 * D_INNER) + d];
    }
    XA[(size_t)m * D_INNER + d] = acc / (1.0f + __expf(-acc));   // SiLU
}

// ---------------------------------------------------------------------------
// Selective scan (fused softplus, dA, recurrence, C-reduce, D-skip, z-gate).
// One thread per (b,d); 16 states in registers. Per-step Bs/Cs (32 floats)
// are staged into LDS by wave 0 using GLOBAL_LOAD_ASYNC_TO_LDS_B32
// (ASYNCcnt-tracked), then published by the workgroup barrier.
// Out may alias DPRE (same-thread read-then-write of identical element).
// ---------------------------------------------------------------------------
__global__ __launch_bounds__(256) void scan_kernel(
    const float* __restrict__ XDBL,  // (M,64): [32:48]=Bs [48:64]=Cs
    const float* __restrict__ DPRE,  // (M,1024): dt@Wdt^T + dtb (pre-softplus)
    const float* __restrict__ XA,    // (M,1024)
    const float* __restrict__ XZ,    // (M,2048): z at +1024
    const float* __restrict__ Alog,  // (1024,16)
    const float* __restrict__ Dp,    // (1024)
    float* __restrict__ YG, int T)
{
    __shared__ float sBC[2 * D_STATE];     // [0:16)=Bs, [16:32)=Cs
    int tid = threadIdx.x;
    int b = blockIdx.x >> 2;
    int d = ((blockIdx.x & 3) << 8) + tid;

    float An[D_STATE];
#pragma unroll
    for (int s = 0; s < D_STATE; ++s)
        An[s] = -__expf(Alog[(size_t)d * D_STATE + s]);
    float dp = Dp[d];
    float h[D_STATE];
#pragma unroll
    for (int s = 0; s < D_STATE; ++s) h[s] = 0.0f;

#if USE_ASYNC_LDS
    // LDS byte address of this lane's slot
    uint32_t lds_dst = (uint32_t)(uintptr_t)&sBC[tid & 31];
#endif

    for (int t = 0; t < T; ++t) {
        size_t row = (size_t)b * T + t;
        if (tid < 2 * D_STATE) {
            const float* gsrc = XDBL + row * 64 + DT_RANK + tid;
#if USE_ASYNC_LDS
            asm volatile("global_load_async_to_lds_b32 %0, %1, off"
                         :: "v"(lds_dst), "v"(gsrc) : "memory");
            asm volatile("s_wait_asynccnt 0x0" ::: "memory");
#else
            sBC[tid] = *gsrc;
#endif
        }
        __syncthreads();

        float pre   = DPRE[row * D_INNER + d];
        float delta = (pre > 20.0f) ? pre : log1pf(__expf(pre));
        float xa = XA[row * D_INNER + d];
        float dx = delta * xa;
        float acc = 0.0f;
#pragma unroll
        for (int s = 0; s < D_STATE; ++s) {
            float da = __expf(delta * An[s]);
            h[s] = da * h[s] + dx * sBC[s];
            acc += h[s] * sBC[D_STATE + s];
        }
        float z = XZ[row * (2 * D_INNER) + D_INNER + d];
        float y = (acc + xa * dp) * (z / (1.0f + __expf(-z)));
        __syncthreads();                 // protect sBC until all lanes done
        YG[row * D_INNER + d] = y;
    }
}

// ---------------------------------------------------------------------------
// Final (B,T,C) token-major -> (T,B,C) output
// ---------------------------------------------------------------------------
__global__ __launch_bounds__(256) void transpose_out(
    const float* __restrict__ Y, float* __restrict__ out)
{
    int tid = blockIdx.x * blockDim.x + threadIdx.x;   // T_LEN*BATCH*D_MODEL
    int c = tid % D_MODEL;
    int b = (tid / D_MODEL) & (BATCH - 1);
    int t = tid / (D_MODEL * BATCH);
    out[tid] = Y[((size_t)b * T_LEN + t) * D_MODEL + c];
}

// ---------------------------------------------------------------------------
static inline void launch_gemm(const float* A, int lda,
                               const float* Bt, int ldbn, int ldbk,
                               const float* bias, float* C, int ldc,
                               int M, int N, int K, int accumulate,
                               hipStream_t stream)
{
    int tiles_m = M / 16, tiles_n64 = N / 64;
    int ntiles = tiles_m * tiles_n64;
    int blocks = (ntiles + 7) / 8;          // 8 waves (tiles) per 256-thr block
    if (ldbk == 1)
        gemm_wmma_f32<true><<<blocks, 256, 0, stream>>>(
            A, lda, Bt, ldbn, ldbk, bias, C, ldc, tiles_m, tiles_n64, K, accumulate);
    else
        gemm_wmma_f32<false><<<blocks, 256, 0, stream>>>(
            A, lda, Bt, ldbn, ldbk, bias, C, ldc, tiles_m, tiles_n64, K, accumulate);
}

extern "C" void kernel_launch(void* const* d_in, const int* in_sizes, int n_in,
                              void* d_out, int out_size, void* d_ws, size_t ws_size,
                              hipStream_t stream)
{
    (void)in_sizes; (void)n_in; (void)out_size; (void)ws_size;

    const float* x     = (const float*)d_in[0];
    const float* Win   = (const float*)d_in[1];
    const float* convw = (const float*)d_in[2];
    const float* convb = (const float*)d_in[3];
    const float* Wx    = (const float*)d_in[4];
    const float* Wdt   = (const float*)d_in[5];
    const float* dtb   = (const float*)d_in[6];
    const float* Alog  = (const float*)d_in[7];
    const float* Dp    = (const float*)d_in[8];
    const float* Wout  = (const float*)d_in[9];
    const float* up_w1 = (const float*)d_in[10];
    const float* up_b1 = (const float*)d_in[11];
    const float* up_w2 = (const float*)d_in[12];
    const float* up_b2 = (const float*)d_in[13];

    float* ws = (float*)d_ws;
    size_t off = 0;
    auto alloc = [&](size_t n) { float* p = ws + off; off += n; return p; };
    float* U0 = alloc(4096ull * 512);
    float* U1 = alloc(2048ull * 512);
    float* U2 = alloc(1024ull * 512);
    float* Y0 = alloc(4096ull * 512);
    float* Y1 = alloc(2048ull * 512);
    float* Y2 = alloc(1024ull * 512);
    float* XZ = alloc(4096ull * 2048);
    float* XA = alloc(4096ull * 1024);
    float* XD = alloc(4096ull * 64);
    float* DE = alloc(4096ull * 1024);      // pre-softplus delta; reused as YG

    pool_transpose<<<(BATCH * D_MODEL * (T_LEN / 4)) / 256, 256, 0, stream>>>(
        x, U0, U1, U2);

    const float* Us[3] = {U0, U1, U2};
    float*       Ys[3] = {Y0, Y1, Y2};
    for (int i = 0; i < 3; ++i) {
        int T = T_LEN >> i;
        int M = BATCH * T;
        // xz = U @ Win^T           (M,2048)
        launch_gemm(Us[i], 512, Win + (size_t)i * 2048 * 512, 512, 1, nullptr,
                    XZ, 2048, M, 2048, 512, 0, stream);
        // depthwise conv + SiLU    (M,1024)
        conv_silu<<<(M * D_INNER) / 256, 256, 0, stream>>>(
            XZ, convw + (size_t)i * D_INNER * 4, convb + (size_t)i * D_INNER,
            XA, T);
        // xdbl = XA @ Wx^T         (M,64)
        launch_gemm(XA, 1024, Wx + (size_t)i * 64 * 1024, 1024, 1, nullptr,
                    XD, 64, M, 64, 1024, 0, stream);
        // delta_pre = dt @ Wdt^T + dtb   (M,1024); dt = XD[:, :32]
        launch_gemm(XD, 64, Wdt + (size_t)i * 1024 * 32, 32, 1,
                    dtb + (size_t)i * 1024, DE, 1024, M, 1024, 32, 0, stream);
        // selective scan + gating  (writes DE in place)
        scan_kernel<<<BATCH * (D_INNER / 256), 256, 0, stream>>>(
            XD, DE, XA, XZ, Alog + (size_t)i * D_INNER * D_STATE,
            Dp + (size_t)i * D_INNER, DE, T);
        // Y_i = YG @ Wout^T        (M,512)
        launch_gemm(DE, 1024, Wout + (size_t)i * 512 * 1024, 1024, 1, nullptr,
                    Ys[i], 512, M, 512, 1024, 0, stream);
    }

    // YM += upsample(YL): Y1[b,2t+k,o] += sum_c Y2[b,t,c]*up_w1[c,o,k] + up_b1[o]
    for (int k = 0; k < 2; ++k)
        launch_gemm(Y2, 512, up_w1 + k, 2, 1024, up_b1,
                    Y1 + (size_t)k * 512, 1024, BATCH * (T_LEN / 4), 512, 512,
                    1, stream);
    // YS += upsample(YM)
    for (int k = 0; k < 2; ++k)
        launch_gemm(Y1, 512, up_w2 + k, 2, 1024, up_b2,
                    Y0 + (size_t)k * 512, 1024, BATCH * (T_LEN / 2), 512, 512,
                    1, stream);

    transpose_out<<<(T_LEN * BATCH * D_MODEL) / 256, 256, 0, stream>>>(
        Y0, (float*)d_out);
}